// BindingPocketGNN_33483565039990
// MI455X (gfx1250) — compile-verified
//
#include <hip/hip_runtime.h>
#include <math.h>

typedef __attribute__((ext_vector_type(2))) float v2f;
typedef __attribute__((ext_vector_type(8))) float v8f;

#define HID 128
#define BN_EPS 1e-5f

// ---------------- degree / dinv ----------------
__global__ void k_init_deg(float* deg, int n) {
  int i = blockIdx.x * blockDim.x + threadIdx.x;
  if (i < n) deg[i] = 1.0f;  // self-loop contributes 1 to every dst degree
}

__global__ void k_deg_edges(const int* __restrict__ dst, int e, float* deg) {
  int i = blockIdx.x * blockDim.x + threadIdx.x;
  if (i < e) atomicAdd(&deg[dst[i]], 1.0f);
}

__global__ void k_dinv(float* deg, int n) {
  int i = blockIdx.x * blockDim.x + threadIdx.x;
  if (i < n) deg[i] = rsqrtf(deg[i]);
}

__global__ void k_zero_stats(float* sums, float* sumsq) {
  int i = threadIdx.x;
  if (i < HID) { sums[i] = 0.0f; sumsq[i] = 0.0f; }
}

// ---------------- GEMM (WMMA f32 16x16x4) ----------------
// hn[row] = (BNReLU(in[row]) @ W) * dinv[row];  agg initialized = hn (self-loop pre-added).
// scale/shift == nullptr -> raw input (layer 1).
__global__ __launch_bounds__(256)
void k_gemm(const float* __restrict__ in, const float* __restrict__ W,
            const float* __restrict__ scale, const float* __restrict__ shift,
            const float* __restrict__ dinv,
            float* __restrict__ hn, float* __restrict__ agg,
            int K, int n) {
  __shared__ float ldsA[16 * (HID + 4)];  // +4 padding breaks bank conflicts
  __shared__ float ldsD[16];
  const int LDA = K + 4;
  const int rowBase = blockIdx.x * 16;
  const int t = threadIdx.x;

  // stage A tile (16 x K) coalesced, fusing prev layer's BN + ReLU
  for (int idx = t; idx < 16 * K; idx += 256) {
    int r = idx / K, c = idx - r * K;
    int row = rowBase + r;
    float v = (row < n) ? in[(size_t)row * K + c] : 0.0f;
    if (scale) v = fmaxf(v * scale[c] + shift[c], 0.0f);
    ldsA[r * LDA + c] = v;
  }
  if (t < 16) ldsD[t] = (rowBase + t < n) ? dinv[rowBase + t] : 0.0f;
  __syncthreads();

  const int wave = t >> 5;       // 8 waves -> 8 column tiles of 16
  const int lane = t & 31;
  const int half = lane >> 4;    // 0: K offset 0,1 ; 1: K offset 2,3
  const int m    = lane & 15;
  const int col  = wave * 16 + m;

  v8f c8 = {};
  for (int k = 0; k < K; k += 4) {
    int kb = k + 2 * half;
    v2f a, b;
    a[0] = ldsA[m * LDA + kb];
    a[1] = ldsA[m * LDA + kb + 1];
    b[0] = W[(size_t)kb * HID + col];
    b[1] = W[(size_t)(kb + 1) * HID + col];
    // D = A(16x4) * B(4x16) + C  ; f32 in, f32 acc
    c8 = __builtin_amdgcn_wmma_f32_16x16x4_f32(false, a, false, b,
                                               (short)0, c8, false, false);
  }

#pragma unroll
  for (int r = 0; r < 8; ++r) {
    int mrow = r + 8 * half;            // C layout: VGPR r holds M=r (lo half) / M=8+r (hi half)
    int row = rowBase + mrow;
    if (row < n) {
      float v = c8[r] * ldsD[mrow];     // pre-scale by dinv[row] -> hn
      size_t o = (size_t)row * HID + col;
      hn[o]  = v;
      agg[o] = v;                       // self-loop term pre-accumulated
    }
  }
}

// ---------------- edge scatter: agg[dst] += hn[src] ----------------
__global__ __launch_bounds__(256)
void k_scatter(const float4* __restrict__ hn4, const int* __restrict__ src,
               const int* __restrict__ dst, float* __restrict__ agg, int E) {
  int tid = blockIdx.x * 256 + threadIdx.x;
  int e = tid >> 5;                      // 32 float4 chunks cover 128 features
  if (e >= E) return;
  int chunk = tid & 31;
  int s = src[e];
  int d = dst[e];
  float4 v = hn4[(size_t)s * (HID / 4) + chunk];
  float* p = agg + (size_t)d * HID + chunk * 4;
  atomicAdd(p + 0, v.x);
  atomicAdd(p + 1, v.y);
  atomicAdd(p + 2, v.z);
  atomicAdd(p + 3, v.w);
}

// ---------------- epilogue: y = agg*dinv + b ; accumulate BN stats ----------------
__global__ __launch_bounds__(128)
void k_epilogue(const float* __restrict__ agg, const float* __restrict__ dinv,
                const float* __restrict__ bias, float* __restrict__ y,
                float* __restrict__ sums, float* __restrict__ sumsq,
                int n, int doStats) {
  int f = threadIdx.x;                  // one feature per thread -> coalesced rows
  int rowBase = blockIdx.x * 128;
  int rend = rowBase + 128; if (rend > n) rend = n;
  float b = bias[f];
  float s = 0.0f, s2 = 0.0f;
  for (int row = rowBase; row < rend; ++row) {
    size_t o = (size_t)row * HID + f;
    float v = agg[o] * dinv[row] + b;
    y[o] = v;
    s += v; s2 += v * v;
  }
  if (doStats) { atomicAdd(&sums[f], s); atomicAdd(&sumsq[f], s2); }
}

// fold BN into per-feature scale/shift for the next GEMM's staged load; reset stats
__global__ void k_bn_finalize(float* sums, float* sumsq,
                              const float* __restrict__ g, const float* __restrict__ bt,
                              float* scale, float* shift, int n) {
  int f = threadIdx.x;
  if (f >= HID) return;
  float inv_n = 1.0f / (float)n;
  float mu  = sums[f] * inv_n;
  float var = sumsq[f] * inv_n - mu * mu;
  float rstd = rsqrtf(var + BN_EPS);
  float sc = rstd * g[f];
  scale[f] = sc;
  shift[f] = bt[f] - mu * sc;
  sums[f] = 0.0f; sumsq[f] = 0.0f;      // ready for next layer
}

// ---------------- FC head: out = relu(y3) @ fcW + fcb ; sigmoid ----------------
__global__ __launch_bounds__(256)
void k_fc(const float* __restrict__ y3, const float* __restrict__ fcW,
          const float* __restrict__ fcb, float* __restrict__ out, int n) {
  int node = (blockIdx.x * 256 + threadIdx.x) >> 5;  // one wave32 per node
  int lane = threadIdx.x & 31;
  if (node >= n) return;
  float acc = 0.0f;
#pragma unroll
  for (int f = lane; f < HID; f += 32) {
    float v = fmaxf(y3[(size_t)node * HID + f], 0.0f);
    acc += v * fcW[f];
  }
#pragma unroll
  for (int off = 16; off > 0; off >>= 1)
    acc += __shfl_down(acc, off, 32);
  if (lane == 0) {
    float o = acc + fcb[0];
    out[node]     = o;                          // output 0: logits
    out[n + node] = 1.0f / (1.0f + expf(-o));   // output 1: sigmoid
  }
}

// ---------------- host launcher ----------------
extern "C" void kernel_launch(void* const* d_in, const int* in_sizes, int n_in,
                              void* d_out, int out_size, void* d_ws, size_t ws_size,
                              hipStream_t stream) {
  const float* x   = (const float*)d_in[0];
  const int*   ei  = (const int*)  d_in[1];
  const float* W1  = (const float*)d_in[2];
  const float* b1  = (const float*)d_in[3];
  const float* g1  = (const float*)d_in[4];
  const float* bt1 = (const float*)d_in[5];
  const float* W2  = (const float*)d_in[6];
  const float* b2  = (const float*)d_in[7];
  const float* g2  = (const float*)d_in[8];
  const float* bt2 = (const float*)d_in[9];
  const float* W3  = (const float*)d_in[10];
  const float* b3  = (const float*)d_in[11];
  const float* fcW = (const float*)d_in[12];
  const float* fcb = (const float*)d_in[13];

  const int IN_DIM = 64;
  const int N = in_sizes[0] / IN_DIM;
  const int E = in_sizes[1] / 2;
  const int* src = ei;        // edge_index[0]
  const int* dst = ei + E;    // edge_index[1]

  // workspace layout
  char* w = (char*)d_ws;
  auto walloc = [&](size_t bytes) -> float* {
    char* p = w;
    w += (bytes + 255) & ~(size_t)255;
    return (float*)p;
  };
  float* dinv  = walloc((size_t)N * 4);
  float* hn    = walloc((size_t)N * HID * 4);
  float* agg   = walloc((size_t)N * HID * 4);
  float* ybuf  = walloc((size_t)N * HID * 4);
  float* sums  = walloc(HID * 4);
  float* sumsq = walloc(HID * 4);
  float* scale = walloc(HID * 4);
  float* shift = walloc(HID * 4);

  const int gemmBlocks = (N + 15) / 16;
  const int scatBlocks = (E * 32 + 255) / 256;
  const int epiBlocks  = (N + 127) / 128;
  const int fcBlocks   = (N * 32 + 255) / 256;

  // degrees (self-loops folded in as init=1), then dinv = deg^{-1/2}
  k_init_deg <<<(N + 255) / 256, 256, 0, stream>>>(dinv, N);
  k_deg_edges<<<(E + 255) / 256, 256, 0, stream>>>(dst, E, dinv);
  k_dinv     <<<(N + 255) / 256, 256, 0, stream>>>(dinv, N);
  k_zero_stats<<<1, 128, 0, stream>>>(sums, sumsq);

  // ---- layer 1 (K=64, raw input) ----
  k_gemm    <<<gemmBlocks, 256, 0, stream>>>(x, W1, nullptr, nullptr, dinv, hn, agg, IN_DIM, N);
  k_scatter <<<scatBlocks, 256, 0, stream>>>((const float4*)hn, src, dst, agg, E);
  k_epilogue<<<epiBlocks, 128, 0, stream>>>(agg, dinv, b1, ybuf, sums, sumsq, N, 1);
  k_bn_finalize<<<1, 128, 0, stream>>>(sums, sumsq, g1, bt1, scale, shift, N);

  // ---- layer 2 (K=128, BN+ReLU fused into A-tile load) ----
  k_gemm    <<<gemmBlocks, 256, 0, stream>>>(ybuf, W2, scale, shift, dinv, hn, agg, HID, N);
  k_scatter <<<scatBlocks, 256, 0, stream>>>((const float4*)hn, src, dst, agg, E);
  k_epilogue<<<epiBlocks, 128, 0, stream>>>(agg, dinv, b2, ybuf, sums, sumsq, N, 1);
  k_bn_finalize<<<1, 128, 0, stream>>>(sums, sumsq, g2, bt2, scale, shift, N);

  // ---- layer 3 (K=128, BN+ReLU fused; no BN stats needed after) ----
  k_gemm    <<<gemmBlocks, 256, 0, stream>>>(ybuf, W3, scale, shift, dinv, hn, agg, HID, N);
  k_scatter <<<scatBlocks, 256, 0, stream>>>((const float4*)hn, src, dst, agg, E);
  k_epilogue<<<epiBlocks, 128, 0, stream>>>(agg, dinv, b3, ybuf, sums, sumsq, N, 0);

  // ---- FC head (ReLU fused into load) + sigmoid ----
  k_fc<<<fcBlocks, 256, 0, stream>>>(ybuf, fcW, fcb, (float*)d_out, N);
}